// Encoder_7249904795876
// MI455X (gfx1250) — compile-verified
//
#include <hip/hip_runtime.h>

// ============================================================================
// MI455X (gfx1250, wave32) implementation of the point-cloud encoder.
// All channel matmuls run on V_WMMA_F32_16X16X4_F32 (exact f32 matrix pipe).
// Each wave computes a 16x32 tile (two independent WMMA accumulator chains)
// for K<=384; a 16x16 tile for the few large-K GEMMs (LDS-bounded).
// Requires ws_size >= ~180 MB (everything is L2-resident on a 192MB-L2 part).
// ============================================================================

typedef __attribute__((ext_vector_type(2))) float v2f;
typedef __attribute__((ext_vector_type(8))) float v8f;

#define WG 128            // 4 waves per block

#define BM_DENSE    0
#define BM_GROUPCAT 1
#define BM_POSREL   2
#define BM_QKPE     3

#define EPI_BIAS   1
#define EPI_RELU   2
#define EPI_BNRELU 4
#define EPI_STATS  8
#define EPI_MAX16  16
#define EPI_RES    32

struct GemmArgs {
  const float* W;            // (M x K), K padded to %4
  const float* bias;         // (M) or null
  const float* Xd;   long xStr;    // dense B operand (K x N)
  const float* xyz;  long xyzStr;  // gather source coords (3 x Nsrc)
  const float* nxyz; long nxyzStr; // group centers (3 x S)
  const float* pts;  long ptsStr;  // gather source features (Cpts x Nsrc)
  const float* q;    long qStr;    // (64 x S)
  const float* km;   long kStr;    // (64 x S)
  const float* pe;   long peStr;   // (64 x S x 16)
  const int*   idx;  long idxStr;  // (S x 16)
  const float* res;  long resStr;  // residual (M x N)
  const float* bnScale;
  const float* bnShift;
  float* statSum;
  float* statSq;
  float* Y; long yStr;
  int M, K, N, Nsrc, S, Cpts, bmode, epi;
};

// ---------------------------------------------------------------------------
// Generic f32 WMMA GEMM. Block = 4 waves stacked in M; all waves share one
// TN-column B panel staged in LDS (panel built per B-operand mode).
// TN==32: each wave runs two independent 16x16 WMMA accumulator chains.
// ---------------------------------------------------------------------------
template <int TNv, int KMAXv>
__global__ __launch_bounds__(WG) void gemm_wmma_kernel(GemmArgs A) {
  __shared__ float Bs[KMAXv * TNv];
  const int tid  = threadIdx.x;
  const int lane = tid & 31;
  const int wave = tid >> 5;
  const int bz   = blockIdx.z;
  const int col0 = blockIdx.x * TNv;
  const int row0 = ((blockIdx.y << 2) + wave) << 4;

  const float* Xd   = A.Xd   + (long)bz * A.xStr;
  const float* xyz  = A.xyz  + (long)bz * A.xyzStr;
  const float* nxyz = A.nxyz + (long)bz * A.nxyzStr;
  const float* pts  = A.pts  + (long)bz * A.ptsStr;
  const float* q    = A.q    + (long)bz * A.qStr;
  const float* km   = A.km   + (long)bz * A.kStr;
  const float* pe   = A.pe   + (long)bz * A.peStr;
  const int*   idx  = A.idx  + (long)bz * A.idxStr;

  // ---- build the K x TN B panel in LDS ----
  const int KN = A.K * TNv;
  for (int t = tid; t < KN; t += WG) {
    const int r = t / TNv;
    const int c = t - r * TNv;
    const int col = col0 + c;
    float v = 0.f;
    if (A.bmode == BM_DENSE) {
      v = Xd[(long)r * A.N + col];
    } else {
      const int s  = col >> 4;                 // group (point)
      const int kk = col & 15;                 // neighbor within group
      const int nbr = idx[(s << 4) + kk];
      if (A.bmode == BM_GROUPCAT) {
        if (r < 3)               v = xyz[r * A.Nsrc + nbr] - nxyz[r * A.S + s];
        else if (r - 3 < A.Cpts) v = pts[(long)(r - 3) * A.Nsrc + nbr];
      } else if (A.bmode == BM_POSREL) {
        if (r < 3)               v = xyz[r * A.S + s] - xyz[r * A.S + nbr];
      } else { // BM_QKPE : q - k_grouped + pe
        v = q[r * A.S + s] - km[r * A.S + nbr] + pe[(((long)r * A.S + s) << 4) + kk];
      }
    }
    Bs[t] = v;
  }
  __syncthreads();   // EXEC all-ones again before the WMMA chain

  // ---- f32 WMMA main loop (K in steps of 4) ----
  v8f accs[2];
  accs[0] = (v8f){0.f, 0.f, 0.f, 0.f, 0.f, 0.f, 0.f, 0.f};
  accs[1] = (v8f){0.f, 0.f, 0.f, 0.f, 0.f, 0.f, 0.f, 0.f};
  const int mrow  = lane & 15;          // A row / B col within tile
  const int khalf = (lane >> 4) << 1;   // lanes 16-31 hold K+2,K+3 (ISA A layout)
  const float* Wr = A.W + (long)(row0 + mrow) * A.K;
  for (int k0 = 0; k0 < A.K; k0 += 4) {
    v2f av;
    av.x = Wr[k0 + khalf];
    av.y = Wr[k0 + khalf + 1];
    v2f bv0;
    bv0.x = Bs[(k0 + khalf) * TNv + mrow];
    bv0.y = Bs[(k0 + khalf + 1) * TNv + mrow];
#if defined(__gfx1250__)
    accs[0] = __builtin_amdgcn_wmma_f32_16x16x4_f32(false, av, false, bv0,
                                                    (short)0, accs[0], false, false);
#else
    (void)av; (void)bv0;
#endif
    if (TNv == 32) {                    // second, independent accumulator chain
      v2f bv1;
      bv1.x = Bs[(k0 + khalf) * TNv + 16 + mrow];
      bv1.y = Bs[(k0 + khalf + 1) * TNv + 16 + mrow];
#if defined(__gfx1250__)
      accs[1] = __builtin_amdgcn_wmma_f32_16x16x4_f32(false, av, false, bv1,
                                                      (short)0, accs[1], false, false);
#else
      (void)bv1;
#endif
    }
  }

  // ---- epilogue (C/D layout: vgpr i -> rows row0+i / row0+8+i) ----
  const int rbase = row0 + ((lane >> 4) << 3);
  float* Y = A.Y + (long)bz * A.yStr;
  const float* res = A.res + (long)bz * A.resStr;
#pragma unroll
  for (int p = 0; p < TNv / 16; ++p) {
    const int ccol = col0 + p * 16 + mrow;
#pragma unroll
    for (int i = 0; i < 8; ++i) {
      const int row = rbase + i;
      float v = accs[p][i];
      if (A.epi & EPI_BIAS)   v += A.bias[row];
      if (A.epi & EPI_BNRELU) v = fmaxf(fmaf(v, A.bnScale[row], A.bnShift[row]), 0.f);
      if (A.epi & EPI_RELU)   v = fmaxf(v, 0.f);
      if (A.epi & EPI_RES)    v += res[(long)row * A.N + ccol];
      if (A.epi & EPI_STATS) {               // channel sum / sumsq for BatchNorm
        float s1 = v, s2 = v * v;
#pragma unroll
        for (int off = 1; off < 16; off <<= 1) {
          s1 += __shfl_xor(s1, off, 32);
          s2 += __shfl_xor(s2, off, 32);
        }
        if (mrow == 0) { atomicAdd(&A.statSum[row], s1); atomicAdd(&A.statSq[row], s2); }
      } else if (A.epi & EPI_MAX16) {        // fused max over the k=16 group
        float m = v;
#pragma unroll
        for (int off = 1; off < 16; off <<= 1) m = fmaxf(m, __shfl_xor(m, off, 32));
        if (mrow == 0) Y[(long)row * A.S + (col0 >> 4) + p] = m;
      } else {
        Y[(long)row * A.N + ccol] = v;
      }
    }
  }
}

// ---------------------------------------------------------------------------
// Furthest point sampling: one block per batch, min-dists in registers,
// LDS argmax (argmax semantics: ties -> lowest index).
// ---------------------------------------------------------------------------
__global__ __launch_bounds__(256) void fps_kernel(const float* xyz, int N, int np, int* fidx) {
  const int b = blockIdx.x;
  const int tid = threadIdx.x;
  const float* xb = xyz + (long)b * 3 * N;
  __shared__ float sv[256];
  __shared__ int   si[256];
  __shared__ int   scur;
  float d[8];
#pragma unroll
  for (int j = 0; j < 8; ++j) d[j] = 1e10f;
  if (tid == 0) scur = 0;
  __syncthreads();
  for (int i = 0; i < np; ++i) {
    const int cur = scur;
    const float cx = xb[cur], cy = xb[N + cur], cz = xb[2 * N + cur];
    float bestv = -1.f; int besti = 0;
#pragma unroll
    for (int j = 0; j < 8; ++j) {
      const int n = j * 256 + tid;
      if (n < N) {
        const float dx = xb[n] - cx, dy = xb[N + n] - cy, dz = xb[2 * N + n] - cz;
        const float dd = dx * dx + dy * dy + dz * dz;
        d[j] = fminf(d[j], dd);
        if (d[j] > bestv) { bestv = d[j]; besti = n; }
      }
    }
    sv[tid] = bestv; si[tid] = besti;
    __syncthreads();
    for (int off = 128; off > 0; off >>= 1) {
      if (tid < off) {
        if (sv[tid + off] > sv[tid] ||
            (sv[tid + off] == sv[tid] && si[tid + off] < si[tid])) {
          sv[tid] = sv[tid + off]; si[tid] = si[tid + off];
        }
      }
      __syncthreads();
    }
    if (tid == 0) { fidx[(long)b * np + i] = cur; scur = si[0]; }
    __syncthreads();
  }
}

// KNN: one thread per query point, register top-16 via unrolled bubble insert.
__global__ void knn_kernel(const float* query, const float* ref, int S, int N,
                           int total, int* outidx) {
  const int t = blockIdx.x * blockDim.x + threadIdx.x;
  if (t >= total) return;
  const int b = t / S, s = t - b * S;
  const float* qb = query + (long)b * 3 * S;
  const float* rb = ref + (long)b * 3 * N;
  const float qx = qb[s], qy = qb[S + s], qz = qb[2 * S + s];
  float bd[16]; int bi[16];
#pragma unroll
  for (int j = 0; j < 16; ++j) { bd[j] = 3.0e38f; bi[j] = 0; }
  for (int n = 0; n < N; ++n) {
    const float dx = rb[n] - qx, dy = rb[N + n] - qy, dz = rb[2 * N + n] - qz;
    const float dd = dx * dx + dy * dy + dz * dz;
    if (dd < bd[15]) {
      bd[15] = dd; bi[15] = n;
#pragma unroll
      for (int j = 15; j >= 1; --j) {
        if (bd[j] < bd[j - 1]) {
          const float tf = bd[j]; bd[j] = bd[j - 1]; bd[j - 1] = tf;
          const int   ti = bi[j]; bi[j] = bi[j - 1]; bi[j - 1] = ti;
        }
      }
    }
  }
#pragma unroll
  for (int j = 0; j < 16; ++j) outidx[(long)t * 16 + j] = bi[j];
}

__global__ void gather_xyz_kernel(const float* xyz, const int* fidx, float* out,
                                  int N, int S, int total) {
  const int t = blockIdx.x * blockDim.x + threadIdx.x;
  if (t >= total) return;
  const int b = t / S, s = t - b * S;
  const int src = fidx[(long)b * S + s];
  const float* xb = xyz + (long)b * 3 * N;
  float* ob = out + (long)b * 3 * S;
  ob[s] = xb[src]; ob[S + s] = xb[N + src]; ob[2 * S + s] = xb[2 * N + src];
}

__global__ void pad_w_kernel(const float* src, float* dst, int Cout, int Cin, int CinP) {
  const int t = blockIdx.x * blockDim.x + threadIdx.x;
  if (t >= Cout * CinP) return;
  const int o = t / CinP, c = t - o * CinP;
  dst[t] = (c < Cin) ? src[(long)o * Cin + c] : 0.f;
}

__global__ void zero_kernel(float* p, int n) {
  const int t = blockIdx.x * blockDim.x + threadIdx.x;
  if (t < n) p[t] = 0.f;
}

// Training-mode BN: scale = g*rsqrt(var+eps); shift = beta - mean*scale.
__global__ void bn_finalize_kernel(const float* sum, const float* sq, const float* g,
                                   const float* beta, float* scale, float* shift,
                                   int C, float invCnt) {
  const int c = blockIdx.x * blockDim.x + threadIdx.x;
  if (c >= C) return;
  const float m  = sum[c] * invCnt;
  const float vr = sq[c] * invCnt - m * m;
  const float sc = g[c] * rsqrtf(vr + 1e-5f);
  scale[c] = sc; shift[c] = beta[c] - m * sc;
}

// softmax over k=16 fused with value gather (+pe) and weighted sum.
__global__ void softmax_agg_kernel(const float* attn, const float* v, const float* pe,
                                   const int* idx, float* agg, int n) {
  const int t = blockIdx.x * blockDim.x + threadIdx.x;
  if (t >= 64 * n) return;
  const int c = t / n, s = t - c * n;
  const float* ar = attn + (long)t * 16;
  float m = ar[0];
#pragma unroll
  for (int j = 1; j < 16; ++j) m = fmaxf(m, ar[j]);
  float e[16]; float sum = 0.f;
#pragma unroll
  for (int j = 0; j < 16; ++j) { e[j] = __expf(ar[j] - m); sum += e[j]; }
  const float inv = 1.0f / sum;
  float acc = 0.f;
#pragma unroll
  for (int j = 0; j < 16; ++j) {
    const int nbr = idx[(long)s * 16 + j];
    acc += e[j] * (v[(long)c * n + nbr] + pe[(long)t * 16 + j]);
  }
  agg[t] = acc * inv;
}

// A3 = concat(l2_xyz, l2_pts) padded to 516 channels, (B,516,512).
__global__ void concat_sa3_kernel(const float* l2xyz, const float* l2pts, float* A3) {
  const int t = blockIdx.x * blockDim.x + threadIdx.x;
  const int total = 16 * 516 * 512;
  if (t >= total) return;
  const int b = t / (516 * 512);
  const int rem = t - b * 516 * 512;
  const int r = rem / 512, s = rem - r * 512;
  float v = 0.f;
  if (r < 3)        v = l2xyz[((long)b * 3 + r) * 512 + s];
  else if (r < 515) v = l2pts[((long)b * 512 + (r - 3)) * 512 + s];
  A3[t] = v;
}

__global__ void colmax_kernel(const float* H, float* l3) {
  const int t = blockIdx.x * blockDim.x + threadIdx.x;
  if (t >= 16 * 1024) return;
  const float* row = H + (long)t * 512;
  float m = row[0];
  for (int s = 1; s < 512; ++s) m = fmaxf(m, row[s]);
  l3[t] = m;
}

// d_out = [l2_xyz (B,3,512)] ++ [concat(l2_pts, broadcast(l3)) (B,1536,512)]
__global__ void write_out_kernel(const float* l2xyz, const float* l2pts,
                                 const float* l3, float* out, long total) {
  const long t = (long)blockIdx.x * blockDim.x + threadIdx.x;
  if (t >= total) return;
  const long t1 = (long)16 * 3 * 512;
  if (t < t1) { out[t] = l2xyz[t]; return; }
  const long u = t - t1;
  const int b = (int)(u / (1536 * 512));
  const int rem = (int)(u - (long)b * 1536 * 512);
  const int c = rem / 512, s = rem - c * 512;
  out[t] = (c < 512) ? l2pts[((long)b * 512 + c) * 512 + s]
                     : l3[(long)b * 1024 + (c - 512)];
}

// ===========================================================================
// Host side
// ===========================================================================
struct Lin { const float* W; const float* b; };
struct BNp { const float* g; const float* b; };
struct TParams { Lin ls, ck, cq, cv, pm1, pm2, am1, am2, le; BNp pbn, abn; };

static inline void launch_gemm(hipStream_t st, const GemmArgs& a, int z) {
  if (a.K <= 384) {   // dual-chain 16x32 tiles (all hot GEMMs)
    dim3 grid((unsigned)(a.N / 32), (unsigned)(a.M / 64), (unsigned)z);
    gemm_wmma_kernel<32, 384><<<grid, dim3(WG), 0, st>>>(a);
  } else {            // large-K (512/516): 16x16 tiles, 33KB LDS
    dim3 grid((unsigned)(a.N / 16), (unsigned)(a.M / 64), (unsigned)z);
    gemm_wmma_kernel<16, 516><<<grid, dim3(WG), 0, st>>>(a);
  }
}

static void run_transformer(hipStream_t st, const TParams& tp,
                            const float* x, float* xout, const float* pos,
                            int n, int inC, int* idxbuf, float* padPosW,
                            float* statSum, float* statSq,
                            float* bnScale, float* bnShift, float* scratch) {
  const int B = 16;
  const long nd  = (long)64 * n;
  const long n16 = (long)n * 16;
  float* x1   = scratch;
  float* q    = x1 + B * nd;
  float* kf   = q + B * nd;
  float* v    = kf + B * nd;
  float* pe   = v + B * nd;            // (B,64,n,16)
  float* agg  = pe + B * nd * 16;
  float* tmp  = agg + B * nd;          // per-batch (64,n*16)
  float* attn = tmp + 64 * n16;        // per-batch (64,n*16)
  float* mid  = attn + 64 * n16;       // per-batch (256,n*16)

  { int total = B * n;
    knn_kernel<<<(total + 255) / 256, 256, 0, st>>>(pos, pos, n, n, total, idxbuf); }
  pad_w_kernel<<<1, 256, 0, st>>>(tp.pm1.W, padPosW, 64, 3, 4);

  { GemmArgs a = {}; a.W = tp.ls.W; a.bias = tp.ls.b; a.M = 64; a.K = inC; a.N = n;
    a.bmode = BM_DENSE; a.Xd = x; a.xStr = (long)inC * n;
    a.epi = EPI_BIAS; a.Y = x1; a.yStr = nd; launch_gemm(st, a, B); }
  { GemmArgs a = {}; a.W = tp.ck.W; a.bias = tp.ck.b; a.M = 64; a.K = 64; a.N = n;
    a.bmode = BM_DENSE; a.Xd = x1; a.xStr = nd; a.epi = EPI_BIAS;
    a.Y = kf; a.yStr = nd; launch_gemm(st, a, B); }
  { GemmArgs a = {}; a.W = tp.cq.W; a.bias = tp.cq.b; a.M = 64; a.K = 64; a.N = n;
    a.bmode = BM_DENSE; a.Xd = x1; a.xStr = nd; a.epi = EPI_BIAS;
    a.Y = q; a.yStr = nd; launch_gemm(st, a, B); }
  { GemmArgs a = {}; a.W = tp.cv.W; a.bias = tp.cv.b; a.M = 64; a.K = 64; a.N = n;
    a.bmode = BM_DENSE; a.Xd = x1; a.xStr = nd; a.epi = EPI_BIAS;
    a.Y = v; a.yStr = nd; launch_gemm(st, a, B); }

  // ---- positional MLP: two-pass training BN (conv bias cancels in BN) ----
  zero_kernel<<<1, 256, 0, st>>>(statSum, 256);
  zero_kernel<<<1, 256, 0, st>>>(statSq, 256);
  { GemmArgs a = {}; a.W = padPosW; a.M = 64; a.K = 4; a.N = (int)n16;
    a.bmode = BM_POSREL; a.xyz = pos; a.xyzStr = (long)3 * n;
    a.idx = idxbuf; a.idxStr = (long)n * 16; a.S = n;
    a.epi = EPI_STATS; a.statSum = statSum; a.statSq = statSq; launch_gemm(st, a, B); }
  bn_finalize_kernel<<<1, 256, 0, st>>>(statSum, statSq, tp.pbn.g, tp.pbn.b,
                                        bnScale, bnShift, 64, 1.0f / ((float)B * n * 16));
  for (int b = 0; b < B; ++b) {
    { GemmArgs a = {}; a.W = padPosW; a.M = 64; a.K = 4; a.N = (int)n16;
      a.bmode = BM_POSREL; a.xyz = pos + (long)b * 3 * n;
      a.idx = idxbuf + (long)b * n * 16; a.S = n;
      a.epi = EPI_BNRELU; a.bnScale = bnScale; a.bnShift = bnShift;
      a.Y = tmp; launch_gemm(st, a, 1); }
    { GemmArgs a = {}; a.W = tp.pm2.W; a.bias = tp.pm2.b; a.M = 64; a.K = 64;
      a.N = (int)n16; a.bmode = BM_DENSE; a.Xd = tmp;
      a.epi = EPI_BIAS; a.Y = pe + (long)b * nd * 16; launch_gemm(st, a, 1); }
  }

  // ---- attention MLP: two-pass BN, (q-k+pe) gathered on the fly ----
  zero_kernel<<<1, 256, 0, st>>>(statSum, 256);
  zero_kernel<<<1, 256, 0, st>>>(statSq, 256);
  { GemmArgs a = {}; a.W = tp.am1.W; a.M = 256; a.K = 64; a.N = (int)n16;
    a.bmode = BM_QKPE; a.q = q; a.qStr = nd; a.km = kf; a.kStr = nd;
    a.pe = pe; a.peStr = nd * 16; a.idx = idxbuf; a.idxStr = (long)n * 16; a.S = n;
    a.epi = EPI_STATS; a.statSum = statSum; a.statSq = statSq; launch_gemm(st, a, B); }
  bn_finalize_kernel<<<1, 256, 0, st>>>(statSum, statSq, tp.abn.g, tp.abn.b,
                                        bnScale, bnShift, 256, 1.0f / ((float)B * n * 16));
  for (int b = 0; b < B; ++b) {
    { GemmArgs a = {}; a.W = tp.am1.W; a.M = 256; a.K = 64; a.N = (int)n16;
      a.bmode = BM_QKPE; a.q = q + (long)b * nd; a.km = kf + (long)b * nd;
      a.pe = pe + (long)b * nd * 16; a.idx = idxbuf + (long)b * n * 16; a.S = n;
      a.epi = EPI_BNRELU; a.bnScale = bnScale; a.bnShift = bnShift;
      a.Y = mid; launch_gemm(st, a, 1); }
    { GemmArgs a = {}; a.W = tp.am2.W; a.bias = tp.am2.b; a.M = 64; a.K = 256;
      a.N = (int)n16; a.bmode = BM_DENSE; a.Xd = mid;
      a.epi = EPI_BIAS; a.Y = attn; launch_gemm(st, a, 1); }
    { int total = 64 * n;
      softmax_agg_kernel<<<(total + 255) / 256, 256, 0, st>>>(
          attn, v + (long)b * nd, pe + (long)b * nd * 16,
          idxbuf + (long)b * n * 16, agg + (long)b * nd, n); }
  }

  // ---- linear_end + residual ----
  { GemmArgs a = {}; a.W = tp.le.W; a.bias = tp.le.b; a.M = inC; a.K = 64; a.N = n;
    a.bmode = BM_DENSE; a.Xd = agg; a.xStr = nd;
    a.epi = EPI_BIAS | EPI_RES; a.res = x; a.resStr = (long)inC * n;
    a.Y = xout; a.yStr = (long)inC * n; launch_gemm(st, a, B); }
}

extern "C" void kernel_launch(void* const* d_in, const int* in_sizes, int n_in,
                              void* d_out, int out_size, void* d_ws, size_t ws_size,
                              hipStream_t stream) {
  (void)n_in; (void)out_size; (void)ws_size;
  const int B = 16;

  // ---------------- parameter mapping (handles both flatten orders) --------
  const float* partial;
  Lin s1c1, s1c2, s2c1, s2c2, s3c1, s3c2; TParams t1 = {}, t2 = {};
  {
    int ii = 0;
    auto f = [&](int i) -> const float* { return (const float*)d_in[i]; };
    auto nextL = [&](Lin& l) { l.W = f(ii++); l.b = f(ii++); };
    auto fillT_ins = [&](TParams& t) {
      nextL(t.ls); nextL(t.ck); nextL(t.cq); nextL(t.cv);
      nextL(t.pm1); t.pbn.g = f(ii++); t.pbn.b = f(ii++); nextL(t.pm2);
      nextL(t.am1); t.abn.g = f(ii++); t.abn.b = f(ii++); nextL(t.am2);
      nextL(t.le);
    };
    auto fillT_sorted = [&](TParams& t) {
      t.abn.b = f(ii++); t.abn.g = f(ii++);
      nextL(t.am1); nextL(t.am2);
      nextL(t.ck); nextL(t.cq); nextL(t.cv);
      nextL(t.le); nextL(t.ls);
      t.pbn.b = f(ii++); t.pbn.g = f(ii++);
      nextL(t.pm1); nextL(t.pm2);
    };
    if (in_sizes[0] == 16 * 3 * 2048) {       // insertion-order flattening
      partial = f(ii++);
      nextL(s1c1); nextL(s1c2); fillT_ins(t1);
      nextL(s2c1); nextL(s2c2); fillT_ins(t2);
      nextL(s3c1); nextL(s3c2);
    } else {                                  // jax pytree sorted-key order
      nextL(s1c1); nextL(s1c2); nextL(s2c1); nextL(s2c2); nextL(s3c1); nextL(s3c2);
      fillT_sorted(t1); fillT_sorted(t2);
      partial = f(ii++);
    }
  }

  // ---------------- workspace bump allocator -------------------------------
  size_t off = 0;
  auto allocB = [&](size_t bytes) -> char* {
    char* p = (char*)d_ws + off; off += (bytes + 255) & ~(size_t)255; return p;
  };
  auto allocF = [&](size_t ne) -> float* { return (float*)allocB(ne * 4); };
  auto allocI = [&](size_t ne) -> int*   { return (int*)allocB(ne * 4); };

  float* w1 = allocF(64 * 8);                 // padded sa1.c1.W
  float* w2 = allocF(256 * 132);              // padded sa2.c1.W
  float* w3 = allocF((size_t)512 * 516);      // padded sa3.c1.W
  float* padPosW1 = allocF(64 * 4);
  float* padPosW2 = allocF(64 * 4);
  int* fidx1 = allocI((size_t)B * 1024);
  int* fidx2 = allocI((size_t)B * 512);
  float* l1_xyz = allocF((size_t)B * 3 * 1024);
  float* l2_xyz = allocF((size_t)B * 3 * 512);
  int* idx_sa1 = allocI((size_t)B * 1024 * 16);
  int* idx_sa2 = allocI((size_t)B * 512 * 16);
  int* idx_t1  = allocI((size_t)B * 1024 * 16);
  int* idx_t2  = allocI((size_t)B * 512 * 16);
  float* l1_pts = allocF((size_t)B * 128 * 1024);
  float* l1b    = allocF((size_t)B * 128 * 1024);
  float* l2_pts = allocF((size_t)B * 512 * 512);
  float* l2b    = allocF((size_t)B * 512 * 512);
  float* l3     = allocF((size_t)B * 1024);
  float* statSum = allocF(256); float* statSq = allocF(256);
  float* bnScale = allocF(256); float* bnShift = allocF(256);
  float* scratch = allocF(28400000);          // shared phase scratch (t1 peak)

  // ================= SA1: fps/knn + gather-fused conv MLP ==================
  pad_w_kernel<<<2, 256, 0, stream>>>(s1c1.W, w1, 64, 6, 8);
  fps_kernel<<<B, 256, 0, stream>>>(partial, 2048, 1024, fidx1);
  { int total = B * 1024;
    gather_xyz_kernel<<<(total + 255) / 256, 256, 0, stream>>>(partial, fidx1, l1_xyz, 2048, 1024, total);
    knn_kernel<<<(total + 255) / 256, 256, 0, stream>>>(l1_xyz, partial, 1024, 2048, total, idx_sa1); }
  float* Hb = scratch;
  for (int b = 0; b < B; ++b) {
    { GemmArgs a = {}; a.W = w1; a.bias = s1c1.b; a.M = 64; a.K = 8; a.N = 1024 * 16;
      a.bmode = BM_GROUPCAT;
      a.xyz = partial + (long)b * 3 * 2048; a.nxyz = l1_xyz + (long)b * 3 * 1024;
      a.pts = partial + (long)b * 3 * 2048; a.Cpts = 3; a.Nsrc = 2048; a.S = 1024;
      a.idx = idx_sa1 + (long)b * 1024 * 16;
      a.epi = EPI_BIAS | EPI_RELU; a.Y = Hb; launch_gemm(stream, a, 1); }
    { GemmArgs a = {}; a.W = s1c2.W; a.bias = s1c2.b; a.M = 128; a.K = 64; a.N = 1024 * 16;
      a.bmode = BM_DENSE; a.Xd = Hb; a.S = 1024;
      a.epi = EPI_BIAS | EPI_MAX16; a.Y = l1_pts + (long)b * 128 * 1024;
      launch_gemm(stream, a, 1); }
  }

  // ================= T1 ====================================================
  run_transformer(stream, t1, l1_pts, l1b, l1_xyz, 1024, 128, idx_t1, padPosW1,
                  statSum, statSq, bnScale, bnShift, scratch);

  // ================= SA2 ===================================================
  pad_w_kernel<<<(256 * 132 + 255) / 256, 256, 0, stream>>>(s2c1.W, w2, 256, 131, 132);
  fps_kernel<<<B, 256, 0, stream>>>(l1_xyz, 1024, 512, fidx2);
  { int total = B * 512;
    gather_xyz_kernel<<<(total + 255) / 256, 256, 0, stream>>>(l1_xyz, fidx2, l2_xyz, 1024, 512, total);
    knn_kernel<<<(total + 255) / 256, 256, 0, stream>>>(l2_xyz, l1_xyz, 512, 1024, total, idx_sa2); }
  for (int b = 0; b < B; ++b) {
    { GemmArgs a = {}; a.W = w2; a.bias = s2c1.b; a.M = 256; a.K = 132; a.N = 512 * 16;
      a.bmode = BM_GROUPCAT;
      a.xyz = l1_xyz + (long)b * 3 * 1024; a.nxyz = l2_xyz + (long)b * 3 * 512;
      a.pts = l1b + (long)b * 128 * 1024; a.Cpts = 128; a.Nsrc = 1024; a.S = 512;
      a.idx = idx_sa2 + (long)b * 512 * 16;
      a.epi = EPI_BIAS | EPI_RELU; a.Y = Hb; launch_gemm(stream, a, 1); }
    { GemmArgs a = {}; a.W = s2c2.W; a.bias = s2c2.b; a.M = 512; a.K = 256; a.N = 512 * 16;
      a.bmode = BM_DENSE; a.Xd = Hb; a.S = 512;
      a.epi = EPI_BIAS | EPI_MAX16; a.Y = l2_pts + (long)b * 512 * 512;
      launch_gemm(stream, a, 1); }
  }

  // ================= T2 ====================================================
  run_transformer(stream, t2, l2_pts, l2b, l2_xyz, 512, 512, idx_t2, padPosW2,
                  statSum, statSq, bnScale, bnShift, scratch);

  // ================= SA3 (group_all) =======================================
  pad_w_kernel<<<((512 * 516) + 255) / 256, 256, 0, stream>>>(s3c1.W, w3, 512, 515, 516);
  float* A3  = scratch;
  float* H3a = A3 + (size_t)B * 516 * 512;
  float* H3b = H3a + (size_t)B * 512 * 512;
  { int total = B * 516 * 512;
    concat_sa3_kernel<<<(total + 255) / 256, 256, 0, stream>>>(l2_xyz, l2b, A3); }
  { GemmArgs a = {}; a.W = w3; a.bias = s3c1.b; a.M = 512; a.K = 516; a.N = 512;
    a.bmode = BM_DENSE; a.Xd = A3; a.xStr = (long)516 * 512;
    a.epi = EPI_BIAS | EPI_RELU; a.Y = H3a; a.yStr = (long)512 * 512;
    launch_gemm(stream, a, B); }
  { GemmArgs a = {}; a.W = s3c2.W; a.bias = s3c2.b; a.M = 1024; a.K = 512; a.N = 512;
    a.bmode = BM_DENSE; a.Xd = H3a; a.xStr = (long)512 * 512;
    a.epi = EPI_BIAS; a.Y = H3b; a.yStr = (long)1024 * 512;
    launch_gemm(stream, a, B); }
  colmax_kernel<<<(16 * 1024 + 255) / 256, 256, 0, stream>>>(H3b, l3);

  // ================= final output ==========================================
  const long total = (long)B * 3 * 512 + (long)B * 1536 * 512;
  write_out_kernel<<<(int)((total + 255) / 256), 256, 0, stream>>>(
      l2_xyz, l2b, l3, (float*)d_out, total);
}